// WaveletLoss_87866440941961
// MI455X (gfx1250) — compile-verified
//
#include <hip/hip_runtime.h>

typedef float v2f __attribute__((ext_vector_type(2)));
typedef float v8f __attribute__((ext_vector_type(8)));
typedef float f4  __attribute__((ext_vector_type(4)));

#define GRID_BLOCKS 2048
#define THREADS     256

// ---------------------------------------------------------------------------
// Kernel 1: streaming Haar-difference partial sums.
// Unit = one float4 column-group of one row-pair = two 2x2 blocks.
// Total units: 32 imgs * 512 row-pairs * 256 col-groups = 4,194,304.
// 2048*256 threads -> 8 units/thread, 64 bytes loaded per unit (b128 x4).
// ---------------------------------------------------------------------------
__global__ __launch_bounds__(THREADS) void haar_partial_kernel(
    const float* __restrict__ pred,
    const float* __restrict__ targ,
    float* __restrict__ partials) {
  const unsigned UNITS = 32u * 512u * 256u;  // 4,194,304
  float s = 0.0f;

  for (unsigned u = blockIdx.x * blockDim.x + threadIdx.x; u < UNITS;
       u += gridDim.x * blockDim.x) {
    unsigned cg  = u & 255u;   // column group (4 floats each)
    unsigned rp  = u >> 8;     // global row-pair index
    unsigned img = rp >> 9;    // batch index (512 row-pairs per image)
    unsigned r   = rp & 511u;  // row-pair within image

    size_t base = (size_t)img * (1024u * 1024u) + (size_t)(2u * r) * 1024u + 4u * cg;

    f4 p0 = *(const f4*)(pred + base);          // row 2r   : a0 b0 a1 b1
    f4 p1 = *(const f4*)(pred + base + 1024);   // row 2r+1 : c0 d0 c1 d1
    f4 t0 = *(const f4*)(targ + base);
    f4 t1 = *(const f4*)(targ + base + 1024);

    f4 e0 = p0 - t0;
    f4 e1 = p1 - t1;

    // block 0: a=e0.x b=e0.y c=e1.x d=e1.y
    float apb = e0.x + e0.y, cpd = e1.x + e1.y;
    float amb = e0.x - e0.y, cmd = e1.x - e1.y;
    s += fabsf(apb - cpd) + fabsf(amb + cmd) + fabsf(amb - cmd);

    // block 1: a=e0.z b=e0.w c=e1.z d=e1.w
    apb = e0.z + e0.w; cpd = e1.z + e1.w;
    amb = e0.z - e0.w; cmd = e1.z - e1.w;
    s += fabsf(apb - cpd) + fabsf(amb + cmd) + fabsf(amb - cmd);
  }

  // wave32 reduction
  #pragma unroll
  for (int off = 16; off > 0; off >>= 1)
    s += __shfl_xor(s, off, 32);

  __shared__ float wsum[THREADS / 32];
  const int lane = threadIdx.x & 31;
  const int wid  = threadIdx.x >> 5;
  if (lane == 0) wsum[wid] = s;
  __syncthreads();

  if (threadIdx.x == 0) {
    float t = 0.0f;
    #pragma unroll
    for (int i = 0; i < THREADS / 32; ++i) t += wsum[i];
    partials[blockIdx.x] = t;
  }
}

// ---------------------------------------------------------------------------
// Kernel 2: one wave32. Reduce `n` partials (n % 64 == 0) with the fp32 WMMA
// V_WMMA_F32_16X16X4_F32 used as an accumulating 64-way adder:
//   A = ones(16x4)  =>  D = A*B + C  adds every element of B (4x16, 64 fp32)
// into the column accumulators of C. Grand total is layout-invariant.
// Row M=0 of the accumulator lives in c[0] on lanes 0..15 (ISA C/D layout).
// ---------------------------------------------------------------------------
__global__ __launch_bounds__(32) void haar_final_kernel(
    const float* __restrict__ partials, float* __restrict__ out, int n) {
  v8f c = {};
  v2f a;
  a.x = 1.0f;
  a.y = 1.0f;
  const int lane = threadIdx.x;  // 0..31, EXEC all ones (no divergence here)

  for (int i = 0; i < n; i += 64) {
    v2f b;
    b.x = partials[i + lane];
    b.y = partials[i + 32 + lane];
    // 8 args: (neg_a, A, neg_b, B, c_mod, C, reuse_a, reuse_b)
    c = __builtin_amdgcn_wmma_f32_16x16x4_f32(
        false, a, false, b, (short)0, c, false, false);
  }

  // Sum the 16 column sums held in c[0] on lanes 0..15 (offsets < 16 stay
  // within the 16-lane half, so lane 0 ends with the row-0 total).
  float v = c[0];
  v += __shfl_xor(v, 8, 32);
  v += __shfl_xor(v, 4, 32);
  v += __shfl_xor(v, 2, 32);
  v += __shfl_xor(v, 1, 32);

  if (lane == 0) {
    // 0.5 Haar factor, /3 losses, /(32*512*512) blocks
    const float scale = 0.5f / (3.0f * 8388608.0f);
    out[0] = v * scale;
  }
}

extern "C" void kernel_launch(void* const* d_in, const int* in_sizes, int n_in,
                              void* d_out, int out_size, void* d_ws, size_t ws_size,
                              hipStream_t stream) {
  const float* pred = (const float*)d_in[0];
  const float* targ = (const float*)d_in[1];
  float* partials   = (float*)d_ws;  // needs GRID_BLOCKS * 4 = 8 KiB

  haar_partial_kernel<<<GRID_BLOCKS, THREADS, 0, stream>>>(pred, targ, partials);
  haar_final_kernel<<<1, 32, 0, stream>>>(partials, (float*)d_out, GRID_BLOCKS);
}